// MPC_68994354643137
// MI455X (gfx1250) — compile-verified
//
#include <hip/hip_runtime.h>

#define T_STEPS 64
#define BSZ     2048
#define NS      32
#define NC      16
#define NN      48
#define LDF     52   // sF row stride (floats): multiple of 4 -> 16B-aligned rows for b128 async

typedef float v2f __attribute__((ext_vector_type(2)));
typedef float v8f __attribute__((ext_vector_type(8)));
typedef int   v4i __attribute__((ext_vector_type(4)));

#if defined(__gfx1250__) && __has_builtin(__builtin_amdgcn_global_load_async_to_lds_b128)
#define HAVE_ASYNC_LDS 1
typedef __attribute__((address_space(1))) v4i g1v4i;  // global int4
typedef __attribute__((address_space(3))) v4i l3v4i;  // LDS int4
__device__ __forceinline__ void async_wait0() {
#if __has_builtin(__builtin_amdgcn_s_wait_asynccnt)
  __builtin_amdgcn_s_wait_asynccnt(0);
#else
  asm volatile("s_wait_asynccnt 0" ::: "memory");
#endif
}
#else
#define HAVE_ASYNC_LDS 0
#endif

// ---- WMMA f32 16x16x4 wrappers (CDNA5, wave32) -----------------------------
__device__ __forceinline__ v8f wmma4(v2f a, v2f b, v8f c) {
  // 8 args: (neg_a, A, neg_b, B, c_mod, C, reuse_a, reuse_b)
  return __builtin_amdgcn_wmma_f32_16x16x4_f32(false, a, false, b, (short)0, c,
                                               false, false);
}

// D(16x16) += A(16xK) * B(Kx16).  A[m][k] = As[m*a_rs + k*a_cs],
// B[k][n] = Bs[k*b_rs + n*b_cs].  mn = lane&15, half = lane>>4.
// ISA layout: A V0 = K(2*half), V1 = K(2*half+1); B analogous.
template <int K>
__device__ __forceinline__ v8f mm_tile(const float* As, int a_rs, int a_cs,
                                       const float* Bs, int b_rs, int b_cs,
                                       v8f acc, int mn, int half) {
#pragma unroll
  for (int k0 = 0; k0 < K; k0 += 4) {
    const int ka = k0 + 2 * half;
    v2f a, b;
    a.x = As[mn * a_rs + ka * a_cs];
    a.y = As[mn * a_rs + (ka + 1) * a_cs];
    b.x = Bs[ka * b_rs + mn * b_cs];
    b.y = Bs[(ka + 1) * b_rs + mn * b_cs];
    acc = wmma4(a, b, acc);
  }
  return acc;
}

// C/D fragment: VGPR r holds row (r + 8*half), col = mn
__device__ __forceinline__ v8f ld_ctile(const float* Cs, int ld, int mn, int half) {
  v8f c;
#pragma unroll
  for (int r = 0; r < 8; ++r) c[r] = Cs[(r + 8 * half) * ld + mn];
  return c;
}
// non-temporal variant for once-streamed global data (keep 192MB L2 clean)
__device__ __forceinline__ v8f ld_ctile_nt(const float* Cs, int ld, int mn, int half) {
  v8f c;
#pragma unroll
  for (int r = 0; r < 8; ++r)
    c[r] = __builtin_nontemporal_load(&Cs[(r + 8 * half) * ld + mn]);
  return c;
}
__device__ __forceinline__ void st_dtile(float* Ds, int ld, v8f d, int mn, int half) {
#pragma unroll
  for (int r = 0; r < 8; ++r) Ds[(r + 8 * half) * ld + mn] = d[r];
}

// ---- One wave32 per batch element: full backward Riccati + forward rollout --
__global__ __launch_bounds__(32) void lqr_kernel(
    const float* __restrict__ x0, const float* __restrict__ Cm,
    const float* __restrict__ cv, const float* __restrict__ Fm,
    const float* __restrict__ fv, float* __restrict__ out,
    float* __restrict__ wsK) {
  const int b    = blockIdx.x;
  const int lane = threadIdx.x;
  const int mn   = lane & 15;
  const int half = lane >> 4;

  __shared__ __align__(16) float sF[NS * LDF];  // Ft [32][48]
  __shared__ float sV[NS * 33];                 // V  [32][32]
  __shared__ float sFtV[NN * 33];               // FtV [48][32]; reused as Vn scratch
  __shared__ float sQ[NN * 49];                 // Q  [48][48]
  __shared__ float sM[NC * 50];                 // Gauss-Jordan augmented [16][49]
  __shared__ __align__(16) float sK[NC * 33];   // [K | k] [16][33]
  __shared__ float sW[NC * 33];                 // Quu@K [16][32]
  __shared__ float sq[NN], sv[NS], sft[NS], sfac[NC], sg[NC], su[NC], sx[NS];

  // V = 0, v = 0
  sv[lane] = 0.f;
  for (int idx = lane; idx < NS * 33; idx += 32) sV[idx] = 0.f;
  __syncthreads();

  // ---------------- backward pass: t = T-1 .. 0 ----------------
  for (int t = T_STEPS - 1; t >= 0; --t) {
    // prefetch next iteration's C and F tiles (global_prefetch_b8)
    if (t > 0) {
      const char* Cn = (const char*)(Cm + ((size_t)(t - 1) * BSZ + b) * (NN * NN));
      for (int off = lane; off < (NN * NN * 4) / 128; off += 32)
        __builtin_prefetch(Cn + off * 128, 0, 0);
      if (t - 1 < T_STEPS - 1) {
        const char* Fn = (const char*)(Fm + ((size_t)(t - 1) * BSZ + b) * (NS * NN));
        for (int off = lane; off < (NS * NN * 4) / 128; off += 32)
          __builtin_prefetch(Fn + off * 128, 0, 0);
      }
    }

    // stage Ft, ft  (zero-padded final step: Fp[T-1]=0, fp[T-1]=0)
    if (t < T_STEPS - 1) {
      const float* Fg = Fm + ((size_t)t * BSZ + b) * (NS * NN);
#if HAVE_ASYNC_LDS
      // async global->LDS (ASYNCcnt path): 384 x b128 chunks, 12 per lane
      for (int idx = lane; idx < (NS * NN) / 4; idx += 32) {
        int s = idx / 12, cc4 = idx - s * 12;  // 12 x 16B chunks per 48-float row
        __builtin_amdgcn_global_load_async_to_lds_b128(
            (g1v4i*)(Fg + idx * 4), (l3v4i*)(sF + s * LDF + cc4 * 4), 0, 0);
      }
      sft[lane] = fv[((size_t)t * BSZ + b) * NS + lane];
      async_wait0();
#else
      for (int idx = lane; idx < NS * NN; idx += 32) {
        int s = idx / NN, i = idx - s * NN;
        sF[s * LDF + i] = Fg[idx];
      }
      sft[lane] = fv[((size_t)t * BSZ + b) * NS + lane];
#endif
    } else {
      for (int idx = lane; idx < NS * LDF; idx += 32) sF[idx] = 0.f;
      sft[lane] = 0.f;
    }
    __syncthreads();

    // FtV[i][s'] = sum_s Ft[s][i] * V[s][s']   (48x32, K=32)
    for (int ti = 0; ti < 3; ++ti)
      for (int tj = 0; tj < 2; ++tj) {
        v8f acc = {0, 0, 0, 0, 0, 0, 0, 0};
        acc = mm_tile<NS>(&sF[ti * 16], 1, LDF, &sV[tj * 16], 33, 1, acc, mn, half);
        st_dtile(&sFtV[(ti * 16) * 33 + tj * 16], 33, acc, mn, half);
      }
    __syncthreads();

    // Q = Ct + FtV @ Ft   (48x48, K=32), accumulator seeded from global Ct (NT)
    const float* Cg = Cm + ((size_t)t * BSZ + b) * (NN * NN);
    for (int ti = 0; ti < 3; ++ti)
      for (int tj = 0; tj < 3; ++tj) {
        v8f acc = ld_ctile_nt(&Cg[(ti * 16) * NN + tj * 16], NN, mn, half);
        acc = mm_tile<NS>(&sFtV[(ti * 16) * 33], 33, 1, &sF[tj * 16], LDF, 1, acc,
                          mn, half);
        st_dtile(&sQ[(ti * 16) * 49 + tj * 16], 49, acc, mn, half);
      }

    // q[i] = ct[i] + sum_s FtV[i][s]*ft[s] + sum_s Ft[s][i]*v[s]
    const float* cg = cv + ((size_t)t * BSZ + b) * NN;
    __syncthreads();
    for (int ib = 0; ib < 2; ++ib) {
      int i = ib * 32 + lane;
      if (i < NN) {
        float acc = cg[i];
        for (int s = 0; s < NS; ++s)
          acc += sFtV[i * 33 + s] * sft[s] + sF[s * LDF + i] * sv[s];
        sq[i] = acc;
      }
    }
    __syncthreads();

    // augmented system [Quu | Qux | qu] (16 x 49)
    for (int idx = lane; idx < NC * 49; idx += 32) {
      int r = idx / 49, cc = idx - r * 49;
      float val;
      if (cc < 16)      val = sQ[(32 + r) * 49 + 32 + cc];
      else if (cc < 48) val = sQ[(32 + r) * 49 + (cc - 16)];
      else              val = sq[32 + r];
      sM[r * 50 + cc] = val;
    }
    __syncthreads();

    // Gauss-Jordan (Quu is SPD -> no pivoting)
    for (int p = 0; p < NC; ++p) {
      float inv = 1.0f / sM[p * 50 + p];
      for (int cc = lane; cc < 49; cc += 32)
        sM[p * 50 + cc] = (cc == p) ? 1.0f : sM[p * 50 + cc] * inv;
      __syncthreads();
      if (lane < NC) sfac[lane] = (lane == p) ? 0.f : sM[lane * 50 + p];
      __syncthreads();
      for (int idx = lane; idx < NC * 49; idx += 32) {
        int r = idx / 49, cc = idx - r * 49;
        sM[r * 50 + cc] -= sfac[r] * sM[p * 50 + cc];
      }
      __syncthreads();
    }

    // [K|k] = -Quu^{-1}[Qux|qu]; keep in LDS and spill to ws for forward pass
    float* wg = wsK + ((size_t)t * BSZ + b) * (NC * 33);
    for (int idx = lane; idx < NC * 33; idx += 32) {
      int r = idx / 33, cc = idx - r * 33;
      float val = (cc < 32) ? -sM[r * 50 + 16 + cc] : -sM[r * 50 + 48];
      sK[idx] = val;
      wg[idx] = val;
    }
    __syncthreads();

    // W = Quu @ K  (16x32, K=16)
    for (int tj = 0; tj < 2; ++tj) {
      v8f acc = {0, 0, 0, 0, 0, 0, 0, 0};
      acc = mm_tile<NC>(&sQ[32 * 49 + 32], 49, 1, &sK[tj * 16], 33, 1, acc, mn, half);
      st_dtile(&sW[tj * 16], 33, acc, mn, half);
    }
    __syncthreads();

    // Vn = Qxx + Qxu@K + K^T@Qux + K^T@W   (32x32)
    float* sVn = sFtV;  // reuse
    for (int ti = 0; ti < 2; ++ti)
      for (int tj = 0; tj < 2; ++tj) {
        v8f acc = ld_ctile(&sQ[(ti * 16) * 49 + tj * 16], 49, mn, half);
        acc = mm_tile<NC>(&sQ[(ti * 16) * 49 + 32], 49, 1, &sK[tj * 16], 33, 1,
                          acc, mn, half);                       // Qxu @ K
        acc = mm_tile<NC>(&sK[ti * 16], 1, 33, &sQ[32 * 49 + tj * 16], 49, 1,
                          acc, mn, half);                       // K^T @ Qux
        acc = mm_tile<NC>(&sK[ti * 16], 1, 33, &sW[tj * 16], 33, 1,
                          acc, mn, half);                       // K^T @ (Quu K)
        st_dtile(&sVn[(ti * 16) * 33 + tj * 16], 33, acc, mn, half);
      }
    __syncthreads();

    // V = 0.5*(Vn + Vn^T);   g = qu + Quu@k
    for (int idx = lane; idx < NS * NS; idx += 32) {
      int i = idx >> 5, j = idx & 31;
      sV[i * 33 + j] = 0.5f * (sVn[i * 33 + j] + sVn[j * 33 + i]);
    }
    if (lane < NC) {
      float g = sq[32 + lane];
      for (int v2 = 0; v2 < NC; ++v2)
        g += sQ[(32 + lane) * 49 + 32 + v2] * sK[v2 * 33 + 32];
      sg[lane] = g;
    }
    __syncthreads();
    // v[x] = qx[x] + sum_u Qxu[x][u]*k[u] + sum_u K[u][x]*g[u]
    {
      float acc = sq[lane];
      for (int u = 0; u < NC; ++u)
        acc += sQ[lane * 49 + 32 + u] * sK[u * 33 + 32] + sK[u * 33 + lane] * sg[u];
      sv[lane] = acc;
    }
    __syncthreads();
  }

  // ---------------- forward rollout: t = 0 .. T-1 ----------------
  sx[lane] = x0[(size_t)b * NS + lane];
  __syncthreads();
  for (int t = 0; t < T_STEPS; ++t) {
    const float* wg = wsK + ((size_t)t * BSZ + b) * (NC * 33);
#if HAVE_ASYNC_LDS
    for (int idx = lane; idx < (NC * 33) / 4; idx += 32)  // 132 x b128 chunks
      __builtin_amdgcn_global_load_async_to_lds_b128(
          (g1v4i*)(wg + idx * 4), (l3v4i*)(sK + idx * 4), 0, 0);
    async_wait0();
#else
    for (int idx = lane; idx < NC * 33; idx += 32) sK[idx] = wg[idx];
#endif
    __syncthreads();
    if (lane < NC) {
      float u = sK[lane * 33 + 32];
      for (int x = 0; x < NS; ++x) u += sK[lane * 33 + x] * sx[x];
      su[lane] = u;
    }
    __syncthreads();
    float* taug = out + ((size_t)t * BSZ + b) * NN;
    __builtin_nontemporal_store(sx[lane], &taug[lane]);
    if (lane < NC) __builtin_nontemporal_store(su[lane], &taug[NS + lane]);
    if (t < T_STEPS - 1) {
      const float* Fg = Fm + ((size_t)t * BSZ + b) * (NS * NN);
      float xn = fv[((size_t)t * BSZ + b) * NS + lane];
      for (int i = 0; i < NS; ++i)
        xn += __builtin_nontemporal_load(&Fg[lane * NN + i]) * sx[i];
      for (int i = 0; i < NC; ++i)
        xn += __builtin_nontemporal_load(&Fg[lane * NN + NS + i]) * su[i];
      __syncthreads();
      sx[lane] = xn;
    }
    __syncthreads();
  }
}

extern "C" void kernel_launch(void* const* d_in, const int* in_sizes, int n_in,
                              void* d_out, int out_size, void* d_ws,
                              size_t ws_size, hipStream_t stream) {
  const float* x0 = (const float*)d_in[0];
  const float* Cm = (const float*)d_in[1];
  const float* cv = (const float*)d_in[2];
  const float* Fm = (const float*)d_in[3];
  const float* fv = (const float*)d_in[4];
  float* out = (float*)d_out;
  float* ws  = (float*)d_ws;  // needs T*B*16*33*4 = ~277 MB for K/k spill
  lqr_kernel<<<BSZ, 32, 0, stream>>>(x0, Cm, cv, Fm, fv, out, ws);
}